// PositionalEmbedding_566935683715
// MI455X (gfx1250) — compile-verified
//
#include <hip/hip_runtime.h>

// Shapes from the reference
#define B_  32
#define C_  256
#define H_  64
#define W_  64
#define NPB (C_ * H_ * W_)     // elements per batch = 1,048,576 (== pe size)
#define ROWS (H_ * W_)         // 4096 pe rows
#define N4  (NPB / 4)          // float4 count per batch = 262,144

typedef __attribute__((ext_vector_type(2))) float v2f;
typedef __attribute__((ext_vector_type(8))) float v8f;

// ---------------------------------------------------------------------------
// Kernel 1: pe[4096, 256] = pos @ W^T + bias  via V_WMMA_F32_16X16X4_F32.
// One wave per 16x16 tile. 256 row-tiles x 16 col-tiles = 4096 tiles.
// A (16x4): built analytically from the coordinate grid.
// B (4x16): W^T, gathered from the 4KB weight matrix.
// C: bias broadcast (each lane owns one output column n -> all 8 accs = b[n]).
// ---------------------------------------------------------------------------
__global__ __launch_bounds__(256) void pe_wmma_kernel(
    const float* __restrict__ Wm,    // [256, 4]
    const float* __restrict__ bias,  // [256]
    float* __restrict__ pe)          // [4096, 256]
{
    const int lane    = threadIdx.x & 31;
    const int waveId  = threadIdx.x >> 5;
    const int tile    = blockIdx.x * 8 + waveId;   // 0..4095
    const int rowTile = tile >> 4;                 // 0..255
    const int colTile = tile & 15;                 // 0..15
    const int half    = lane >> 4;                 // 0: K=0,1   1: K=2,3
    const int m       = lane & 15;

    // ---- A matrix (16x4 F32): row r of pos = [jj, ii, 1-jj, 1-ii]
    const int   r  = rowTile * 16 + m;                      // 0..4095
    const float jj = (float)(r & 63) * (1.0f / 63.0f);
    const float ii = (float)(r >> 6) * (1.0f / 63.0f);
    v2f a;
    a.x = half ? (1.0f - jj) : jj;   // K = 2*half + 0
    a.y = half ? (1.0f - ii) : ii;   // K = 2*half + 1

    // ---- B matrix (4x16 F32): B[K][n] = W[n][K]
    const int n  = colTile * 16 + m;     // output column (channel c)
    const int k0 = half * 2;
    v2f bm;
    bm.x = Wm[n * 4 + k0];
    bm.y = Wm[n * 4 + k0 + 1];

    // ---- C: bias (constant down each lane's column)
    const float bn = bias[n];
    v8f c;
#pragma unroll
    for (int v = 0; v < 8; ++v) c[v] = bn;

    // D = A x B + C   -> v_wmma_f32_16x16x4_f32
    v8f d = __builtin_amdgcn_wmma_f32_16x16x4_f32(
        /*neg_a=*/false, a, /*neg_b=*/false, bm,
        /*c_mod=*/(short)0, c, /*reuse_a=*/false, /*reuse_b=*/false);

    // ---- Store: lane holds rows baseRow..baseRow+7 of column n
    const int baseRow = rowTile * 16 + half * 8;
#pragma unroll
    for (int v = 0; v < 8; ++v)
        pe[(baseRow + v) * C_ + n] = d[v];
}

// ---------------------------------------------------------------------------
// Kernel 2: out[b*NPB + k] = x[b*NPB + k] + pe[k].
// One thread per float4 of pe; pe value loaded once and reused across all 32
// batches (pe = 4 MiB, resident in the 192 MB L2). All accesses are b128,
// fully coalesced (wave32 x 16B = 512B contiguous).
// ---------------------------------------------------------------------------
__global__ __launch_bounds__(256) void add_pe_kernel(
    const float* __restrict__ x,
    const float* __restrict__ pe,
    float* __restrict__ out)
{
    const int k4 = blockIdx.x * 256 + threadIdx.x;   // 0..N4-1
    const float4* __restrict__ xin = reinterpret_cast<const float4*>(x);
    const float4* __restrict__ pin = reinterpret_cast<const float4*>(pe);
    float4* __restrict__ o = reinterpret_cast<float4*>(out);

    const float4 p = pin[k4];

#pragma unroll 4
    for (int b = 0; b < B_; ++b) {
        const int idx = b * N4 + k4;
        float4 xv = xin[idx];
        float4 ov;
        ov.x = xv.x + p.x;
        ov.y = xv.y + p.y;
        ov.z = xv.z + p.z;
        ov.w = xv.w + p.w;
        o[idx] = ov;
    }
}

extern "C" void kernel_launch(void* const* d_in, const int* in_sizes, int n_in,
                              void* d_out, int out_size, void* d_ws, size_t ws_size,
                              hipStream_t stream) {
    const float* x    = (const float*)d_in[0];  // [32,256,64,64]
    const float* Wm   = (const float*)d_in[1];  // [256,4]
    const float* bias = (const float*)d_in[2];  // [256]
    float* out = (float*)d_out;
    float* pe  = (float*)d_ws;                  // 4096*256 floats = 4 MiB

    // Kernel 1: 4096 tiles, 8 waves (one tile each) per 256-thread block.
    pe_wmma_kernel<<<ROWS * (C_ / 16) / 8 / 16, 256, 0, stream>>>(Wm, bias, pe);
    // 4096 tiles / 8 waves-per-block = 512 blocks
    // (expression above: 4096*16/8/16 = 512)

    // Kernel 2: 262,144 float4 threads -> 1024 blocks of 256.
    add_pe_kernel<<<N4 / 256, 256, 0, stream>>>(x, pe, out);
}